// KalmanFilterCost_48009144435191
// MI455X (gfx1250) — compile-verified
//
#include <hip/hip_runtime.h>
#include <math.h>

// Problem constants (from reference setup_inputs): B=8192, T=128, D=32, U=16
#define NB   8192
#define TT   128
#define DD   32
#define UU   16
#define NSTEP (TT - 1)          // 127 scan steps
#define LOG_2PI 1.8378770664093453f

// Workspace layout (in floats)
#define OFS_FT   0                        // F^T  B-operand swizzled: 1024
#define OFS_BMT  1024                     // Bm^T B-operand swizzled: 512
#define OFS_LD   1536                     // logdet[t]: 128 (padded)
#define OFS_KT   1664                     // K^T_t  swizzled: 127*1024
#define OFS_LT   (OFS_KT + NSTEP*1024)    // Linv^T_t swizzled: 127*1024
#define OFS_PART (OFS_LT + NSTEP*1024)    // per-block quad partials: 512

typedef __attribute__((ext_vector_type(2))) float v2f;
typedef __attribute__((ext_vector_type(8))) float v8f;

// ---------------------------------------------------------------------------
// Kernel 1: sequential Riccati recursion (batch-independent!).
// One block of 1024 threads; thread (i,j) owns element (i,j) of 32x32 mats.
// Exports K^T / Linv^T already swizzled into WMMA B-operand per-lane layout:
//   flat idx: j=idx&1, L=(idx>>1)&31, nt=(idx>>6)&1, c=idx>>7
//   value = M[nt*16 + (L&15)][4c + 2*(L>>4) + j]
// ---------------------------------------------------------------------------
__global__ __launch_bounds__(1024) void riccati_precompute(
    const float* __restrict__ A, const float* __restrict__ Bm,
    const float* __restrict__ noise_p, float* __restrict__ ws)
{
  __shared__ float Fm[32][32];
  __shared__ float Gm[32][32];
  __shared__ float covC[32][32];
  __shared__ float covP[32][32];
  __shared__ float SL[32][32];
  __shared__ float Lm[32][32];
  __shared__ float Li[32][32];
  __shared__ float Si[32][32];   // also reused as tmp
  __shared__ float Km[32][32];

  const int tid = threadIdx.x;
  const int i = tid >> 5;
  const int j = tid & 31;
  const float noise = noise_p[0];

  // F = A + I ; G = noise * Bm Bm^T ; cov0 = 0
  Fm[i][j] = A[i*32 + j] + ((i == j) ? 1.0f : 0.0f);
  {
    float g = 0.f;
    for (int k = 0; k < UU; ++k) g += Bm[i*UU + k] * Bm[j*UU + k];
    Gm[i][j] = noise * g;
  }
  covC[i][j] = 0.f;
  __syncthreads();

  // Export F^T and Bm^T in B-operand swizzle
  {
    const int jj = tid & 1, L = (tid >> 1) & 31, nt = (tid >> 6) & 1, c = tid >> 7;
    const int n = nt*16 + (L & 15), k = 4*c + 2*(L >> 4) + jj;
    ws[OFS_FT + tid] = Fm[n][k];                 // (X @ F^T): B[k][n] = F[n][k]
    if (tid < 512) ws[OFS_BMT + tid] = Bm[n*UU + k]; // here k in [0,16)
  }

  for (int t = 0; t < NSTEP; ++t) {
    // tmp = covC @ F^T   (tmp in Si)
    float acc = 0.f;
    for (int k = 0; k < 32; ++k) acc += covC[i][k] * Fm[j][k];
    Si[i][j] = acc;
    __syncthreads();
    // covP = F @ tmp + G ; SL = covP + noise*I
    acc = Gm[i][j];
    for (int k = 0; k < 32; ++k) acc += Fm[i][k] * Si[k][j];
    covP[i][j] = acc;
    SL[i][j] = acc + ((i == j) ? noise : 0.f);
    __syncthreads();

    // In-place Cholesky, one barrier per column.
    // Phase c reads only column c (+diag c,c) and writes only columns > c
    // of SL, plus column c of the separate Lm array -> no read/write overlap.
    for (int c = 0; c < 32; ++c) {
      const float rs = 1.0f / sqrtf(SL[c][c]);
      if (j == c && i >= c) Lm[i][c] = SL[i][c] * rs;
      if (j > c && i >= j)  SL[i][j] -= (SL[i][c] * rs) * (SL[j][c] * rs);
      __syncthreads();
    }

    // logdet = 2*sum(log diag L)
    if (tid == 0) {
      float ld = 0.f;
      for (int d = 0; d < 32; ++d) ld += logf(Lm[d][d]);
      ws[OFS_LD + t] = 2.0f * ld;
    }
    // Linv: column-parallel forward substitution (threads 0..31 = columns)
    if (tid < 32) {
      for (int r = 0; r < 32; ++r) {
        float s = (r == tid) ? 1.0f : 0.0f;
        for (int k = tid; k < r; ++k) s -= Lm[r][k] * Li[k][tid];
        Li[r][tid] = (r >= tid) ? s / Lm[r][r] : 0.0f;
      }
    }
    __syncthreads();
    // Sinv = Linv^T @ Linv
    acc = 0.f;
    for (int k = 0; k < 32; ++k) acc += Li[k][i] * Li[k][j];
    Si[i][j] = acc;
    __syncthreads();
    // K = covP @ Sinv
    acc = 0.f;
    for (int k = 0; k < 32; ++k) acc += covP[i][k] * Si[k][j];
    Km[i][j] = acc;
    __syncthreads();
    // covC' = covP - K @ covP ; export K^T, Linv^T swizzled
    acc = covP[i][j];
    for (int k = 0; k < 32; ++k) acc -= Km[i][k] * covP[k][j];
    covC[i][j] = acc;
    {
      const int jj = tid & 1, L = (tid >> 1) & 31, nt = (tid >> 6) & 1, c = tid >> 7;
      const int n = nt*16 + (L & 15), k = 4*c + 2*(L >> 4) + jj;
      ws[OFS_KT + t*1024 + tid] = Km[n][k];   // B[k][n] = K[n][k]
      ws[OFS_LT + t*1024 + tid] = Li[n][k];   // B[k][n] = Linv[n][k]
    }
    __syncthreads();
  }
}

// ---------------------------------------------------------------------------
// Kernel 2: batched mean recursion. One wave (32 lanes) per block, 16 batch
// rows per wave. All matrix math via V_WMMA_F32_16X16X4_F32 (FP32 in/out).
// A-operand (16x4 f32): lane L, comp j -> (m = L&15, k = 2*(L>>4)+j)
// C/D (16x16 f32):      lane L, vgpr r -> (m = r + 8*(L>>4), n = L&15)
// ---------------------------------------------------------------------------
__global__ __launch_bounds__(32) void kalman_main(
    const float* __restrict__ x, const float* __restrict__ u,
    const float* __restrict__ ws, float* __restrict__ partial)
{
  __shared__ float sb[16 * 32];           // 2 KB tile transpose buffer
  const int lane = threadIdx.x;
  const int h   = lane >> 4;              // half-wave select
  const int m15 = lane & 15;
  const int b0  = blockIdx.x * 16;

  // Preload constant B-operands: F^T (8 K-chunks x 2 N-tiles), Bm^T (4 x 2)
  v2f ft[8][2], bt[4][2];
#pragma unroll
  for (int c = 0; c < 8; ++c)
#pragma unroll
    for (int nt = 0; nt < 2; ++nt)
      ft[c][nt] = *(const v2f*)&ws[OFS_FT + c*128 + nt*64 + lane*2];
#pragma unroll
  for (int c = 0; c < 4; ++c)
#pragma unroll
    for (int nt = 0; nt < 2; ++nt)
      bt[c][nt] = *(const v2f*)&ws[OFS_BMT + c*128 + nt*64 + lane*2];

  // X_corr(t=0) = x[:,0], loaded directly in A-operand layout
  v2f xc[8];
#pragma unroll
  for (int c = 0; c < 8; ++c) {
    const int k = 4*c + 2*h;
    xc[c] = *(const v2f*)&x[(size_t)(b0 + m15) * TT * DD + k];
  }

  float q = 0.f;
  for (int t = 0; t < NSTEP; ++t) {
    // per-step gain / whitening matrices (L2-resident, coalesced float2)
    v2f kt[8][2], lt[8][2];
    const float* wk = &ws[OFS_KT + t*1024];
    const float* wl = &ws[OFS_LT + t*1024];
#pragma unroll
    for (int c = 0; c < 8; ++c)
#pragma unroll
      for (int nt = 0; nt < 2; ++nt) {
        kt[c][nt] = *(const v2f*)&wk[c*128 + nt*64 + lane*2];
        lt[c][nt] = *(const v2f*)&wl[c*128 + nt*64 + lane*2];
      }
    // u_prev tile in A-operand layout (K = 16 control dims -> 4 chunks)
    v2f ua[4];
#pragma unroll
    for (int c = 0; c < 4; ++c) {
      const int k = 4*c + 2*h;
      ua[c] = *(const v2f*)&u[((size_t)(b0 + m15) * TT + t) * UU + k];
    }
    // x_obs(t+1) tile directly in C/D layout
    v8f xo[2];
#pragma unroll
    for (int nt = 0; nt < 2; ++nt)
#pragma unroll
      for (int r = 0; r < 8; ++r)
        xo[nt][r] = x[((size_t)(b0 + r + 8*h) * TT + (t + 1)) * DD + nt*16 + m15];

    // X_pred = X_corr @ F^T + U @ Bm^T
    v8f xp[2];
#pragma unroll
    for (int nt = 0; nt < 2; ++nt) {
      v8f a = {0.f, 0.f, 0.f, 0.f, 0.f, 0.f, 0.f, 0.f};
#pragma unroll
      for (int c = 0; c < 8; ++c)
        a = __builtin_amdgcn_wmma_f32_16x16x4_f32(false, xc[c], false, ft[c][nt],
                                                  (short)0, a, false, false);
#pragma unroll
      for (int c = 0; c < 4; ++c)
        a = __builtin_amdgcn_wmma_f32_16x16x4_f32(false, ua[c], false, bt[c][nt],
                                                  (short)0, a, false, false);
      xp[nt] = a;
    }

    // innov = x_obs - X_pred  (C/D layout) ; transpose to A layout via LDS
    v8f in_[2];
#pragma unroll
    for (int nt = 0; nt < 2; ++nt) in_[nt] = xo[nt] - xp[nt];
#pragma unroll
    for (int nt = 0; nt < 2; ++nt)
#pragma unroll
      for (int r = 0; r < 8; ++r)
        sb[(r + 8*h) * 32 + nt*16 + m15] = in_[nt][r];
    __syncthreads();
    v2f ia[8];
#pragma unroll
    for (int c = 0; c < 8; ++c)
      ia[c] = *(const v2f*)&sb[m15 * 32 + 4*c + 2*h];
    __syncthreads();

    // X_corr' = X_pred + innov @ K^T
    v8f xn[2];
#pragma unroll
    for (int nt = 0; nt < 2; ++nt) {
      v8f a = xp[nt];
#pragma unroll
      for (int c = 0; c < 8; ++c)
        a = __builtin_amdgcn_wmma_f32_16x16x4_f32(false, ia[c], false, kt[c][nt],
                                                  (short)0, a, false, false);
      xn[nt] = a;
    }
    // Y = innov @ Linv^T ; quad += |Y|^2
#pragma unroll
    for (int nt = 0; nt < 2; ++nt) {
      v8f a = {0.f, 0.f, 0.f, 0.f, 0.f, 0.f, 0.f, 0.f};
#pragma unroll
      for (int c = 0; c < 8; ++c)
        a = __builtin_amdgcn_wmma_f32_16x16x4_f32(false, ia[c], false, lt[c][nt],
                                                  (short)0, a, false, false);
#pragma unroll
      for (int r = 0; r < 8; ++r) q += a[r] * a[r];
    }

    // X_corr' : C/D layout -> A layout via LDS for the next step
#pragma unroll
    for (int nt = 0; nt < 2; ++nt)
#pragma unroll
      for (int r = 0; r < 8; ++r)
        sb[(r + 8*h) * 32 + nt*16 + m15] = xn[nt][r];
    __syncthreads();
#pragma unroll
    for (int c = 0; c < 8; ++c)
      xc[c] = *(const v2f*)&sb[m15 * 32 + 4*c + 2*h];
    __syncthreads();
  }

  // deterministic per-block partial: wave reduce then single store
#pragma unroll
  for (int o = 16; o > 0; o >>= 1) q += __shfl_xor(q, o, 32);
  if (lane == 0) partial[blockIdx.x] = q;
}

// ---------------------------------------------------------------------------
// Kernel 3: deterministic final reduction to the scalar NLL.
// ---------------------------------------------------------------------------
__global__ void kalman_finalize(const float* __restrict__ ws,
                                float* __restrict__ out)
{
  float s = 0.f;
  for (int t = 0; t < NSTEP; ++t)
    s += 0.5f * ((float)DD * LOG_2PI + ws[OFS_LD + t]);
  float qs = 0.f;
  for (int w = 0; w < NB / 16; ++w) qs += ws[OFS_PART + w];
  out[0] = s + 0.5f * qs / (float)NB;
}

extern "C" void kernel_launch(void* const* d_in, const int* in_sizes, int n_in,
                              void* d_out, int out_size, void* d_ws, size_t ws_size,
                              hipStream_t stream) {
  const float* x     = (const float*)d_in[0];  // (8192,128,32) f32
  const float* u     = (const float*)d_in[1];  // (8192,128,16) f32
  const float* noise = (const float*)d_in[2];  // scalar f32
  const float* A     = (const float*)d_in[3];  // (32,32) f32
  const float* Bm    = (const float*)d_in[4];  // (32,16) f32
  float* ws  = (float*)d_ws;
  float* out = (float*)d_out;

  riccati_precompute<<<1, 1024, 0, stream>>>(A, Bm, noise, ws);
  kalman_main<<<NB / 16, 32, 0, stream>>>(x, u, ws, ws + OFS_PART);
  kalman_finalize<<<1, 1, 0, stream>>>(ws, out);
}